// ControlDeltaBlock_66073776882307
// MI455X (gfx1250) — compile-verified
//
#include <hip/hip_runtime.h>
#include <hip/hip_bf16.h>

// ---------------------------------------------------------------------------
// ControlDeltaBlock on MI455X (gfx1250), wave32, WMMA f32 16x16x4 +
// async global->LDS staging (ASYNCcnt path).
//   B=2, T=1024, D_IN=1024, D_MEM=256, VOCAB=32000, PHASE=8, MARGIN=4
// ---------------------------------------------------------------------------

typedef __attribute__((ext_vector_type(2))) float v2f;
typedef __attribute__((ext_vector_type(8))) float v8f;

__device__ __forceinline__ v8f wmma_f32(v2f a, v2f b, v8f c) {
  // D = A(16x4 f32) * B(4x16 f32) + C(16x16 f32)
  return __builtin_amdgcn_wmma_f32_16x16x4_f32(
      /*neg_a=*/false, a, /*neg_b=*/false, b,
      /*c_mod=*/(short)0, c, /*reuse_a=*/false, /*reuse_b=*/false);
}

// Async 16B global -> LDS copy (GLOBAL_LOAD_ASYNC_TO_LDS_B128, ASYNCcnt).
// dst must be a pointer into LDS (generic ptr low 32 bits = LDS byte offset).
__device__ __forceinline__ void async_copy_b128(void* lds_dst, const void* gsrc) {
  unsigned dst = (unsigned)(unsigned long long)lds_dst;
  unsigned long long src = (unsigned long long)gsrc;
  asm volatile("global_load_async_to_lds_b128 %0, %1, off"
               :: "v"(dst), "v"(src) : "memory");
}
__device__ __forceinline__ void async_wait0() {
  asm volatile("s_wait_asynccnt 0x0" ::: "memory");
}

#define ROWS 2048        // B*T
#define DIN  1024
#define DM   256
#define VOCAB 32000
#define ASTR 1036        // LDS row stride for 16x1024 A tile (16B aligned, bank-spread)
#define SSTR 260         // LDS row stride for 16x256 A tile

// ---------------------------------------------------------------------------
// Kernel 1: four projections  out = cf @ W^T + b  (sigmoid for beta/lam)
// grid.x = 128 m-tiles, 128 threads (4 waves). A tile staged in LDS (async).
// ---------------------------------------------------------------------------
extern "C" __global__ __launch_bounds__(128)
void proj_kernel(const float* __restrict__ cf,
                 const float* __restrict__ Wk, const float* __restrict__ bk,
                 const float* __restrict__ Wv, const float* __restrict__ bv,
                 const float* __restrict__ Wb, const float* __restrict__ bb,
                 const float* __restrict__ Wl, const float* __restrict__ bl,
                 float* __restrict__ ko, float* __restrict__ vo,
                 float* __restrict__ bo, float* __restrict__ lo)
{
  extern __shared__ float As[];
  const int tid = threadIdx.x;
  const int m0  = blockIdx.x * 16;
  // stage 16 x 1024 rows of cf into LDS via async b128 transfers
  for (int i4 = tid; i4 < 16 * 256; i4 += 128) {
    int rr = i4 >> 8, cc = (i4 & 255) << 2;
    async_copy_b128(&As[rr * ASTR + cc],
                    &cf[(size_t)(m0 + rr) * DIN + cc]);
  }
  async_wait0();
  __syncthreads();

  const int lane = tid & 31, wv = tid >> 5;
  const int r = lane & 15, half = lane >> 4;
  for (int s = 0; s < 16; ++s) {
    const int tile = wv + 4 * s;     // 0..63 = 4 matrices x 16 n-tiles
    const int mat  = tile >> 4;
    const int n0   = (tile & 15) * 16;
    const float *W, *bia; float* out;
    if      (mat == 0) { W = Wk; bia = bk; out = ko; }
    else if (mat == 1) { W = Wv; bia = bv; out = vo; }
    else if (mat == 2) { W = Wb; bia = bb; out = bo; }
    else               { W = Wl; bia = bl; out = lo; }

    v8f c = {};
    const float* wrow = W + (size_t)(n0 + r) * DIN + 2 * half;
    const float* arow = As + r * ASTR + 2 * half;
    for (int k0 = 0; k0 < DIN; k0 += 4)
      c = wmma_f32(*(const v2f*)(arow + k0), *(const v2f*)(wrow + k0), c);

    const float bias = bia[n0 + r];
#pragma unroll
    for (int g = 0; g < 8; ++g) {
      const int row = m0 + g + 8 * half;
      float val = c[g] + bias;
      if (mat >= 2) val = 1.0f / (1.0f + __expf(-val));   // sigmoid
      out[(size_t)row * DM + n0 + r] = val;
    }
  }
}

// ---------------------------------------------------------------------------
// Kernel 2: sequential gated delta-rule scan. One workgroup per batch,
// M (256x256 f32, column-major in LDS) resident for all 1024 steps.
// Normalization folded algebraically: with s = ||k_raw||^2, inv = 1/max(s^.5,eps):
//   read       = (M @ k_raw) * inv
//   corr       = (v - read) * beta
//   M         <- M*lam_row + (corr*inv) (x) k_raw
//   summary    = lam*read + corr * (s*inv^2)
// ---------------------------------------------------------------------------
extern "C" __global__ __launch_bounds__(1024)
void scan_kernel(const float* __restrict__ mem_in,
                 const float* __restrict__ kb, const float* __restrict__ vb,
                 const float* __restrict__ bbuf, const float* __restrict__ lbuf,
                 float* __restrict__ sraw, float* __restrict__ out_mem)
{
  extern __shared__ float sm[];
  float* Ms    = sm;               // 65536 : M[j*256 + i] (col-major)
  float* kvec  = sm + 65536;       // 256 (raw k)
  float* vvec  = kvec + 256;
  float* betav = vvec + 256;
  float* lamv  = betav + 256;
  float* corrv = lamv + 256;       // corr * inv
  float* rpart = corrv + 256;      // 1024
  float* red   = rpart + 1024;     // 8
  float* scal  = red + 8;          // 2 : inv, s*inv^2

  const int tid = threadIdx.x;
  const int b   = blockIdx.x;
  const int i   = tid & 255;       // memory row this thread owns
  const int jb  = (tid >> 8) * 64; // 64-column slice

  for (int idx = tid; idx < DM * DM; idx += 1024) {
    int ii = idx >> 8, jj = idx & 255;
    Ms[jj * DM + ii] = mem_in[b * DM * DM + idx];
  }
  __syncthreads();

  const size_t base = (size_t)b * 1024 * DM;
  for (int t = 0; t < 1024; ++t) {
    const size_t row = base + (size_t)t * DM;
    if (tid < 256) {
      const float kx = kb[row + tid];
      kvec[tid]  = kx;
      vvec[tid]  = vb[row + tid];
      betav[tid] = bbuf[row + tid];
      lamv[tid]  = lbuf[row + tid];
      float x = kx * kx;                       // ||k||^2 reduction (8 waves)
      for (int m = 16; m >= 1; m >>= 1) x += __shfl_xor(x, m, 32);
      if ((tid & 31) == 0) red[tid >> 5] = x;
    }
    __syncthreads();
    if (tid == 0) {
      float s = 0.f;
      for (int w = 0; w < 8; ++w) s += red[w];
      const float inv = 1.0f / fmaxf(sqrtf(s), 1e-12f);
      scal[0] = inv;               // 1/max(||k||,1e-12)
      scal[1] = s * inv * inv;     // ||k_norm||^2
    }
    __syncthreads();

    // rawdot[i] = sum_j M[i][j] * k_raw[j]  (each thread: 64-wide slice)
    float p = 0.f;
    for (int jj = 0; jj < 64; ++jj) {
      const int j = jb + jj;
      p += Ms[j * DM + i] * kvec[j];
    }
    rpart[tid] = p;
    __syncthreads();

    const float inv = scal[0];
    if (tid < 256) {
      const float rd = (rpart[tid] + rpart[tid + 256] +
                        rpart[tid + 512] + rpart[tid + 768]) * inv;
      const float corr = (vvec[tid] - rd) * betav[tid];
      corrv[tid] = corr * inv;               // pre-scaled for the raw-k update
      sraw[row + tid] = lamv[tid] * rd + corr * scal[1];
    }
    __syncthreads();

    // M[i][j] = M[i][j]*lam[i] + (corr[i]*inv)*k_raw[j]
    const float lam_i = lamv[i], corrs_i = corrv[i];
    for (int jj = 0; jj < 64; ++jj) {
      const int j = jb + jj;
      Ms[j * DM + i] = Ms[j * DM + i] * lam_i + corrs_i * kvec[j];
    }
    __syncthreads();
  }

  for (int idx = tid; idx < DM * DM; idx += 1024) {
    int ii = idx >> 8, jj = idx & 255;
    out_mem[b * DM * DM + idx] = Ms[jj * DM + ii];
  }
}

// ---------------------------------------------------------------------------
// Kernel 3: summary = tanh(sraw @ Wsum^T + bsum), [2048,256]x[256,256]
// ---------------------------------------------------------------------------
extern "C" __global__ __launch_bounds__(128)
void summary_kernel(const float* __restrict__ sraw, const float* __restrict__ Ws,
                    const float* __restrict__ bs, float* __restrict__ outS)
{
  extern __shared__ float As[];
  const int tid = threadIdx.x;
  const int m0  = blockIdx.x * 16;
  for (int i4 = tid; i4 < 16 * 64; i4 += 128) {
    int rr = i4 >> 6, cc = (i4 & 63) << 2;
    async_copy_b128(&As[rr * SSTR + cc],
                    &sraw[(size_t)(m0 + rr) * DM + cc]);
  }
  async_wait0();
  __syncthreads();

  const int lane = tid & 31, wv = tid >> 5;
  const int r = lane & 15, half = lane >> 4;
  for (int s = 0; s < 4; ++s) {
    const int n0 = (wv + 4 * s) * 16;
    v8f c = {};
    const float* wrow = Ws + (size_t)(n0 + r) * DM + 2 * half;
    const float* arow = As + r * SSTR + 2 * half;
    for (int k0 = 0; k0 < DM; k0 += 4)
      c = wmma_f32(*(const v2f*)(arow + k0), *(const v2f*)(wrow + k0), c);
    const float bias = bs[n0 + r];
#pragma unroll
    for (int g = 0; g < 8; ++g) {
      const int row = m0 + g + 8 * half;
      outS[(size_t)row * DM + n0 + r] = tanhf(c[g] + bias);
    }
  }
}

// ---------------------------------------------------------------------------
// Kernel 4: vocab heads (logit_bias, winner_logits): [2048,256]x[256,32000]
// grid = (500 n-chunks of 64, 128 m-tiles, 2 heads); 4 waves, 1 tile each.
// Write-bound: 524 MB of f32 logits => ~22.5us floor at 23.3 TB/s.
// ---------------------------------------------------------------------------
extern "C" __global__ __launch_bounds__(128)
void heads_kernel(const float* __restrict__ summ,
                  const float* __restrict__ W0, const float* __restrict__ b0,
                  float* __restrict__ o0,
                  const float* __restrict__ W1, const float* __restrict__ b1,
                  float* __restrict__ o1)
{
  extern __shared__ float As[];
  const float* W   = blockIdx.z ? W1 : W0;
  const float* bia = blockIdx.z ? b1 : b0;
  float*       out = blockIdx.z ? o1 : o0;

  const int tid = threadIdx.x;
  const int m0  = blockIdx.y * 16;
  const int n0  = blockIdx.x * 64 + (tid >> 5) * 16;

  for (int i4 = tid; i4 < 16 * 64; i4 += 128) {
    int rr = i4 >> 6, cc = (i4 & 63) << 2;
    async_copy_b128(&As[rr * SSTR + cc],
                    &summ[(size_t)(m0 + rr) * DM + cc]);
  }
  async_wait0();
  __syncthreads();

  const int lane = tid & 31;
  const int r = lane & 15, half = lane >> 4;
  v8f c = {};
  const float* wrow = W + (size_t)(n0 + r) * DM + 2 * half;
  const float* arow = As + r * SSTR + 2 * half;
  for (int k0 = 0; k0 < DM; k0 += 4) {
    if ((k0 & 15) == 0)
      __builtin_prefetch(wrow + k0 + 64, 0, 1);   // global_prefetch_b8
    c = wmma_f32(*(const v2f*)(arow + k0), *(const v2f*)(wrow + k0), c);
  }
  const float bias = bia[n0 + r];
#pragma unroll
  for (int g = 0; g < 8; ++g) {
    const int row = m0 + g + 8 * half;
    out[(size_t)row * VOCAB + n0 + r] = c[g] + bias;
  }
}

// ---------------------------------------------------------------------------
// Kernel 5: tiny heads (phase 8, margin 4) — one thread per row.
// ---------------------------------------------------------------------------
extern "C" __global__ __launch_bounds__(256)
void small_heads_kernel(const float* __restrict__ summ,
                        const float* __restrict__ Wph, const float* __restrict__ bph,
                        const float* __restrict__ Wmar, const float* __restrict__ bmar,
                        float* __restrict__ oph, float* __restrict__ omar)
{
  const int row = blockIdx.x * 256 + threadIdx.x;   // 0..2047
  const float* s = summ + (size_t)row * DM;
  for (int p = 0; p < 8; ++p) {
    float acc = bph[p];
    const float* w = Wph + p * DM;
    for (int k = 0; k < DM; ++k) acc += s[k] * w[k];
    oph[(size_t)row * 8 + p] = acc;
  }
  for (int m = 0; m < 4; ++m) {
    float acc = bmar[m];
    const float* w = Wmar + m * DM;
    for (int k = 0; k < DM; ++k) acc += s[k] * w[k];
    omar[(size_t)row * 4 + m] = acc;
  }
}

// ---------------------------------------------------------------------------
extern "C" void kernel_launch(void* const* d_in, const int* in_sizes, int n_in,
                              void* d_out, int out_size, void* d_ws, size_t ws_size,
                              hipStream_t stream)
{
  const float* cf   = (const float*)d_in[0];
  const float* mem  = (const float*)d_in[1];
  const float* Wk   = (const float*)d_in[2];  const float* bk   = (const float*)d_in[3];
  const float* Wv   = (const float*)d_in[4];  const float* bv   = (const float*)d_in[5];
  const float* Wb   = (const float*)d_in[6];  const float* bb   = (const float*)d_in[7];
  const float* Wl   = (const float*)d_in[8];  const float* bl   = (const float*)d_in[9];
  const float* Wsum = (const float*)d_in[10]; const float* bsum = (const float*)d_in[11];
  const float* Wlb  = (const float*)d_in[12]; const float* blb  = (const float*)d_in[13];
  const float* Wph  = (const float*)d_in[14]; const float* bph  = (const float*)d_in[15];
  const float* Wwin = (const float*)d_in[16]; const float* bwin = (const float*)d_in[17];
  const float* Wmar = (const float*)d_in[18]; const float* bmar = (const float*)d_in[19];

  float* out      = (float*)d_out;
  float* out_mem  = out;                   // [2,256,256]   = 131072
  float* out_sum  = out + 131072;          // [2,1024,256]  = 524288
  float* out_lb   = out + 655360;          // [2,1024,32000]
  float* out_ph   = out + 66191360;        // [2,1024,8]
  float* out_win  = out + 66207744;        // [2,1024,32000]
  float* out_mar  = out + 131743744;       // [2,1024,4]

  // Scratch lives inside the logit_bias region; heads_kernel (which writes
  // logit_bias) runs after every consumer of the scratch. Deterministic.
  float* kbuf    = out_lb;
  float* vbuf    = kbuf + 524288;
  float* betabuf = vbuf + 524288;
  float* lambuf  = betabuf + 524288;
  float* srawbuf = lambuf + 524288;

  const size_t shProj = (size_t)16 * ASTR * sizeof(float);                 // ~66 KB
  const size_t shScan = (size_t)(65536 + 5 * 256 + 1024 + 8 + 2) * sizeof(float); // ~271 KB
  const size_t shTile = (size_t)16 * SSTR * sizeof(float);                 // ~16.6 KB

  proj_kernel<<<dim3(128), dim3(128), shProj, stream>>>(
      cf, Wk, bk, Wv, bv, Wb, bb, Wl, bl, kbuf, vbuf, betabuf, lambuf);

  scan_kernel<<<dim3(2), dim3(1024), shScan, stream>>>(
      mem, kbuf, vbuf, betabuf, lambuf, srawbuf, out_mem);

  summary_kernel<<<dim3(128), dim3(128), shTile, stream>>>(
      srawbuf, Wsum, bsum, out_sum);

  heads_kernel<<<dim3(VOCAB / 64, 128, 2), dim3(128), shTile, stream>>>(
      out_sum, Wlb, blb, out_lb, Wwin, bwin, out_win);

  small_heads_kernel<<<dim3(8), dim3(256), 0, stream>>>(
      out_sum, Wph, bph, Wmar, bmar, out_ph, out_mar);
}